// LocallyConnectedGC_52587579572679
// MI455X (gfx1250) — compile-verified
//
#include <hip/hip_runtime.h>

typedef __attribute__((ext_vector_type(2))) float v2f;
typedef __attribute__((ext_vector_type(8))) float v8f;

#define NNODES 208
#define KHOP 4
#define ROWS_PER_BLOCK 16
#define LDS_STRIDE 212          // 208 + 4: keeps 16B alignment for b128 LDS stores,
                                // and 212*m mod 64 = 20*m mod 64 is distinct for m=0..15
#define THREADS 416             // 13 waves: one wave per 16-column tile (13*16 = 208)

__global__ __launch_bounds__(THREADS)
void LocallyConnectedGC_kernel(const float* __restrict__ x,
                               const float* __restrict__ W,
                               const float* __restrict__ bias,
                               const float* __restrict__ S,
                               float* __restrict__ out)
{
    __shared__ float xs[ROWS_PER_BLOCK * LDS_STRIDE];

    const int tid  = threadIdx.x;
    const int wave = tid >> 5;            // 0..12 -> output column tile
    const int lane = tid & 31;
    const long long row_base = (long long)blockIdx.x * ROWS_PER_BLOCK;

    // ---- cooperative b128 staging of x tile [16 rows x 208 nodes] into LDS ----
    // 16 * 52 = 832 float4s, 416 threads -> exactly 2 vector loads per thread.
    {
        const float4* __restrict__ x4 = (const float4*)x;
        #pragma unroll
        for (int i = tid; i < ROWS_PER_BLOCK * (NNODES / 4); i += THREADS) {
            const int r  = i / (NNODES / 4);
            const int c4 = i - r * (NNODES / 4);
            const float4 v = x4[(row_base + r) * (NNODES / 4) + c4];
            *(float4*)&xs[r * LDS_STRIDE + c4 * 4] = v;
        }
    }

    // ---- build 6 B-fragments: masked kernel (S*W) for this column tile ----
    // Band: output node m only sees input nodes within ring-distance <= 4,
    // so K range is [col0-4, col0+19] (24 values -> 6 WMMA K=4 steps).
    const int col0  = wave * 16;
    const int n     = col0 + (lane & 15);   // output node handled by this lane
    const int khalf = (lane >> 4) * 2;      // lanes 16-31 hold K = k0+2, k0+3

    v2f Bfrag[6];
    #pragma unroll
    for (int s = 0; s < 6; ++s) {
        #pragma unroll
        for (int j = 0; j < 2; ++j) {
            int k = col0 - KHOP + 4 * s + khalf + j;
            if (k < 0) k += NNODES;
            if (k >= NNODES) k -= NNODES;
            const int idx = k * NNODES + n;
            Bfrag[s][j] = S[idx] * W[idx];  // L2-resident, tiny
        }
    }

    __syncthreads();

    // ---- 6 chained V_WMMA_F32_16X16X4_F32 accumulations ----
    v8f acc = {};
    const int m = lane & 15;                // A-matrix row (bt-row within tile)
    #pragma unroll
    for (int s = 0; s < 6; ++s) {
        v2f Afrag;
        #pragma unroll
        for (int j = 0; j < 2; ++j) {
            int k = col0 - KHOP + 4 * s + khalf + j;
            if (k < 0) k += NNODES;
            if (k >= NNODES) k -= NNODES;
            Afrag[j] = xs[m * LDS_STRIDE + k];
        }
        // 8 args: (neg_a, A, neg_b, B, c_mod, C, reuse_a, reuse_b)
        acc = __builtin_amdgcn_wmma_f32_16x16x4_f32(
            false, Afrag, false, Bfrag[s], (short)0, acc, false, false);
    }

    // ---- bias + store; D layout: VGPR r -> rows r (lanes 0-15) / r+8 (lanes 16-31) ----
    const float bv   = bias[n];
    const int rhalf  = (lane >> 4) * 8;
    #pragma unroll
    for (int r = 0; r < 8; ++r) {
        out[(row_base + r + rhalf) * NNODES + n] = acc[r] + bv;
    }
}

extern "C" void kernel_launch(void* const* d_in, const int* in_sizes, int n_in,
                              void* d_out, int out_size, void* d_ws, size_t ws_size,
                              hipStream_t stream)
{
    (void)in_sizes; (void)n_in; (void)d_ws; (void)ws_size;
    const float* x = (const float*)d_in[0];   // [64, 2048, 208] f32
    const float* W = (const float*)d_in[1];   // [208, 208] f32
    const float* b = (const float*)d_in[2];   // [208] f32
    const float* S = (const float*)d_in[3];   // [208, 208] f32 band mask
    float* out     = (float*)d_out;           // [64, 2048, 208] f32

    const long long total_rows = (long long)out_size / NNODES;   // 131072
    const int grid = (int)(total_rows / ROWS_PER_BLOCK);         // 8192

    LocallyConnectedGC_kernel<<<grid, THREADS, 0, stream>>>(x, W, b, S, out);
}